// FCN_81621558493619
// MI455X (gfx1250) — compile-verified
//
#include <hip/hip_runtime.h>
#include <hip/hip_bf16.h>

typedef __attribute__((ext_vector_type(2))) float v2f;
typedef __attribute__((ext_vector_type(8))) float v8f;

// GEMV  y = [relu](W * x + b)  using V_WMMA_F32_16X16X4_F32.
// Workgroup: 256 threads (8 waves, wave32). 16 output rows per workgroup,
// K dimension split across the 8 waves; cross-wave reduction in LDS.
// A-tile (16x4 f32) per-lane layout == one b64 load from the W row:
//   lane L (0..15):  W[row=L][k], W[row=L][k+1]
//   lane L (16..31): W[row=L-16][k+2], W[row=L-16][k+3]
// B is the broadcast vector (every column = x), same per-lane addressing.
template <bool RELU>
__global__ __launch_bounds__(256) void gemv_wmma(const float* __restrict__ W,
                                                 const float* __restrict__ bias,
                                                 const float* __restrict__ xin,
                                                 float* __restrict__ yout,
                                                 int fin, int fout) {
    __shared__ float red[16];
    const int tid = threadIdx.x;
    if (tid < 16) red[tid] = 0.0f;
    __syncthreads();

    const int lane = tid & 31;
    const int wave = tid >> 5;
    const int rb   = blockIdx.x * 16;

    int row = rb + (lane & 15);
    if (row >= fout) row = fout - 1;          // tail clamp (stores are guarded)
    const int koff = (lane >> 4) * 2;         // 0 for lanes 0-15, 2 for lanes 16-31

    const float* __restrict__ Wrow = W + (size_t)row * (size_t)fin;
    const int kslice = fin >> 3;              // fin = 8192 -> 1024 per wave
    const int k0 = wave * kslice;
    const int k1 = k0 + kslice;

    v8f acc0 = {0.f,0.f,0.f,0.f,0.f,0.f,0.f,0.f};
    v8f acc1 = {0.f,0.f,0.f,0.f,0.f,0.f,0.f,0.f};

    #pragma unroll 4
    for (int k = k0; k < k1; k += 8) {
        v2f a0 = *(const v2f*)(Wrow + k + koff);
        v2f b0 = *(const v2f*)(xin  + k + koff);
        v2f a1 = *(const v2f*)(Wrow + k + 4 + koff);
        v2f b1 = *(const v2f*)(xin  + k + 4 + koff);
        acc0 = __builtin_amdgcn_wmma_f32_16x16x4_f32(false, a0, false, b0,
                                                     (short)0, acc0, false, false);
        acc1 = __builtin_amdgcn_wmma_f32_16x16x4_f32(false, a1, false, b1,
                                                     (short)0, acc1, false, false);
    }
    acc0 += acc1;

    // D layout: lane 0 VGPR v -> M=v (N=0), lane 16 VGPR v -> M=8+v. All N equal.
    if ((lane & 15) == 0) {
        const int mbase = (lane >> 4) * 8;
        #pragma unroll
        for (int v = 0; v < 8; ++v)
            atomicAdd(&red[mbase + v], acc0[v]);   // ds_add_f32
    }
    __syncthreads();

    if (tid < 16) {
        const int r = rb + tid;
        if (r < fout) {
            float val = red[tid] + bias[r];
            if (RELU) val = fmaxf(val, 0.0f);
            yout[r] = val;
        }
    }
}

// Sigma[j] = 0.1 * sum_{d=0..9} Q[j+d],  j < n  (n = 2*size+1)
__global__ void sigma_kernel(const float* __restrict__ Q,
                             float* __restrict__ Sigma, int n) {
    const int j = blockIdx.x * blockDim.x + threadIdx.x;
    if (j >= n) return;
    float s = 0.0f;
    #pragma unroll
    for (int d = 0; d < 10; ++d) s += Q[j + d];
    Sigma[j] = 0.1f * s;
}

// Per-temperature interpolation + trapezoid integrals.
// t_j = Wc*(0.02*j - 1);  pos = size*(1 + x_i*(0.02*j - 1));
// I1 = (Wc/50) * sum'' Sint_j * g_j
// I2 = (Wc/50) * x_i * sum'' t_j * Sint_j * g_j     (sum'' = half end weights)
__global__ void integral_kernel(const float* __restrict__ x,
                                const int* __restrict__ WcPtr,
                                const float* __restrict__ Sigma,
                                float* __restrict__ out, int size) {
    const int i = blockIdx.x * blockDim.x + threadIdx.x;
    if (i >= size) return;
    const float Wc = (float)(*WcPtr);
    const float xi = x[i];
    const float fs = (float)size;
    const int   imax = 2 * size - 1;

    float s1 = 0.0f, s2 = 0.0f;
    for (int j = 0; j <= 100; ++j) {
        const float tn  = 0.02f * (float)j - 1.0f;     // in [-1, 1]
        const float tj  = Wc * tn;
        float pos = fs * (1.0f + xi * tn);
        int i0 = (int)floorf(pos);
        i0 = min(max(i0, 0), imax);
        const float frac = pos - (float)i0;
        const float lo = Sigma[i0];
        const float hi = Sigma[i0 + 1];
        const float Sint = lo + (hi - lo) * frac;
        const float eu  = __expf(tj);
        const float den = eu + 1.0f;
        const float g   = eu / (den * den);
        const float f   = Sint * g;
        const float wgt = (j == 0 || j == 100) ? 0.5f : 1.0f;
        s1 += wgt * f;
        s2 += wgt * tj * f;
    }
    const float c = Wc * 0.02f;                        // Wc / 50
    out[i]        =  c * s1;
    out[size + i] = -(c * xi * s2);
}

extern "C" void kernel_launch(void* const* d_in, const int* in_sizes, int n_in,
                              void* d_out, int out_size, void* d_ws, size_t ws_size,
                              hipStream_t stream) {
    const float* x  = (const float*)d_in[0];
    const int*   Wc = (const int*)  d_in[1];
    const float* W1 = (const float*)d_in[2];
    const float* b1 = (const float*)d_in[3];
    const float* W2 = (const float*)d_in[4];
    const float* b2 = (const float*)d_in[5];
    const float* W3 = (const float*)d_in[6];
    const float* b3 = (const float*)d_in[7];
    float* out = (float*)d_out;

    const int size = in_sizes[0];   // 8192
    const int f1   = in_sizes[3];   // 8192
    const int f2   = in_sizes[5];   // 8192
    const int f3   = in_sizes[7];   // 2*size + 10

    float* h1    = (float*)d_ws;    // [f1]
    float* h2    = h1 + f1;         // [f2]
    float* Q     = h2 + f2;         // [f3]
    float* Sigma = Q  + f3;         // [f3 - 9] = 2*size+1
    const int nS = f3 - 9;

    gemv_wmma<true ><<<dim3((f1 + 15) / 16), 256, 0, stream>>>(W1, b1, x,  h1, size, f1);
    gemv_wmma<true ><<<dim3((f2 + 15) / 16), 256, 0, stream>>>(W2, b2, h1, h2, f1,   f2);
    gemv_wmma<false><<<dim3((f3 + 15) / 16), 256, 0, stream>>>(W3, b3, h2, Q,  f2,   f3);
    sigma_kernel   <<<dim3((nS + 255) / 256), 256, 0, stream>>>(Q, Sigma, nS);
    integral_kernel<<<dim3((size + 255) / 256), 256, 0, stream>>>(x, Wc, Sigma, out, size);
}